// ConditionalFeedForward_26955214750384
// MI455X (gfx1250) — compile-verified
//
#include <hip/hip_runtime.h>
#include <hip/hip_bf16.h>

// MoE SwiGLU FFN, top-2 routing, fp32 WMMA (V_WMMA_F32_16X16X4_F32) with
// TDM (tensor_load_to_lds) double-buffered weight staging.
// Memory-bound (~1 GB @ 23.3 TB/s ~= 42us); weights DMA'd by the Tensor Data
// Mover with native LDS pad (68-float pitch) while WMMAs run on the other buffer.

constexpr int T = 2048, A = 2, E = 8, I = 4096, D = 2048;
constexpr int MAXP = T * A;        // worst-case pairs per expert
constexpr int BK = 64;             // K chunk staged through LDS
constexpr int LSTR = 68;           // LDS row pitch (floats): 64 data + 4 pad

typedef float v2f __attribute__((ext_vector_type(2)));
typedef float v4f __attribute__((ext_vector_type(4)));
typedef float v8f __attribute__((ext_vector_type(8)));
typedef unsigned int u32x4 __attribute__((ext_vector_type(4)));
typedef int i32x4 __attribute__((ext_vector_type(4)));
typedef int i32x8 __attribute__((ext_vector_type(8)));

// Low 32 bits of a flat pointer into the LDS aperture == LDS byte address.
__device__ inline unsigned lds_off(const void* p) {
  return (unsigned)(uintptr_t)p;
}

// TDM: DMA a tile of `rows` x 64 fp32 from global (row stride `stride_elems`
// elements) into LDS at `lds_addr`, inserting 4 pad DWORDs after every 64
// data DWORDs (-> LSTR=68 pitch). Issue from ONE wave only (EXEC ignored).
__device__ inline void tdm_load_tile(unsigned lds_addr, const float* g,
                                     unsigned rows, unsigned long long stride_elems) {
  unsigned long long ga = (unsigned long long)(uintptr_t)g;
  u32x4 g0;
  g0[0] = 1u;                                        // count=1 (valid user D#)
  g0[1] = lds_addr;                                  // lds_addr [63:32]
  g0[2] = (unsigned)(ga & 0xFFFFFFFFu);              // global_addr lo
  g0[3] = (unsigned)((ga >> 32) & 0x01FFFFFFu)       // global_addr hi [120:96]
        | 0x80000000u;                               // type=2 ('10' in [127:126])
  const unsigned td0 = 64u;                          // tensor_dim0 (exact, no OOB)
  const unsigned td1 = rows;                         // tensor_dim1
  i32x8 g1;
  g1[0] = (int)((2u << 16)                           // data_size = 4B
              | (1u << 20)                           // pad_enable
              | (5u << 22)                           // pad_interval: 64 DWORDs
              | (3u << 25));                         // pad_amount:   4 DWORDs
  g1[1] = (int)((td0 & 0xFFFFu) << 16);              // [63:48] tensor_dim0 lo
  g1[2] = (int)((td0 >> 16) | ((td1 & 0xFFFFu) << 16));
  g1[3] = (int)((td1 >> 16) | (64u << 16));          // [127:112] tile_dim0 = 64
  g1[4] = (int)rows;                                 // [143:128] tile_dim1
  g1[5] = (int)(unsigned)(stride_elems & 0xFFFFFFFFull);   // dim0_stride lo
  g1[6] = (int)(unsigned)((stride_elems >> 32) & 0xFFFFull);
  g1[7] = 0;
  i32x4 gz = {0, 0, 0, 0};
#if defined(__clang_major__) && __clang_major__ >= 23
  i32x8 gz8 = {0, 0, 0, 0, 0, 0, 0, 0};
  __builtin_amdgcn_tensor_load_to_lds(g0, g1, gz, gz, gz8, 0);
#else
  __builtin_amdgcn_tensor_load_to_lds(g0, g1, gz, gz, 0);
#endif
}

// ---------------- routing ----------------

__global__ void moe_zero(int* counts) {
  if (threadIdx.x < E) counts[threadIdx.x] = 0;
}

__global__ void moe_route(const int* __restrict__ eidx, int* counts, int* pairs) {
  int p = blockIdx.x * blockDim.x + threadIdx.x;
  if (p < T * A) {
    int e = eidx[p];
    int pos = atomicAdd(&counts[e], 1);
    pairs[e * MAXP + pos] = p;   // pid = t*A + a
  }
}

// ---------------- GEMM1: h[pid,i] = silu(x.w1^T) * (x.w3^T) ----------------
// grid: (I/64, MAXP/16, E), block 256 (8 waves).
// waves 0-3: w1 subtiles, waves 4-7: w3 subtiles. Weights via TDM, 2 buffers.

__global__ __launch_bounds__(256)
void moe_gemm1(const float* __restrict__ x, const float* __restrict__ w1,
               const float* __restrict__ w3, const int* __restrict__ counts,
               const int* __restrict__ pairs, float* __restrict__ h) {
  const int e   = blockIdx.z;
  const int cnt = counts[e];
  const int m0  = blockIdx.y * 16;
  if (m0 >= cnt) return;
  const int i0  = blockIdx.x * 64;

  __shared__ float xs[2][16 * LSTR];
  __shared__ float ws[2][2][64 * LSTR];   // [buf][mat]
  __shared__ float hbuf[4 * 32 * 9];

  const int tid  = threadIdx.x;
  const int lane = tid & 31;
  const int wave = tid >> 5;
  const int half = lane >> 4;
  const int mat  = wave >> 2;      // 0 -> w1, 1 -> w3
  const int nsub = wave & 3;
  const int nlo  = lane & 15;

  const int xr = tid >> 4;
  const int xc = (tid & 15) * 4;
  const float* xrow = nullptr;
  if (m0 + xr < cnt) {
    int pid = pairs[e * MAXP + m0 + xr];
    xrow = x + (size_t)(pid >> 1) * D;   // token = pid / A (A==2)
  }

  const float* w1base = w1 + ((size_t)e * I + i0) * D;
  const float* w3base = w3 + ((size_t)e * I + i0) * D;

  // prologue: stage chunk 0
  if (tid < 32) {
    tdm_load_tile(lds_off(&ws[0][0][0]), w1base, 64, D);
    tdm_load_tile(lds_off(&ws[0][1][0]), w3base, 64, D);
  }
  {
    v4f xv = {};
    if (xrow) xv = *(const v4f*)(xrow + xc);
    *(v4f*)(&xs[0][xr * LSTR + xc]) = xv;
  }

  v8f c = {};
  const int nIter = D / BK;
  for (int it = 0; it < nIter; ++it) {
    const int cur = it & 1;
    if (tid < 32) __builtin_amdgcn_s_wait_tensorcnt(0);
    __syncthreads();   // cur buffers ready (TDM done, x stores drained)

    const bool pref = (it + 1 < nIter);
    v4f xv = {};
    if (pref) {
      const int k0n = (it + 1) * BK;
      if (tid < 32) {
        tdm_load_tile(lds_off(&ws[cur ^ 1][0][0]), w1base + k0n, 64, D);
        tdm_load_tile(lds_off(&ws[cur ^ 1][1][0]), w3base + k0n, 64, D);
      }
      if (xrow) xv = *(const v4f*)(xrow + k0n + xc);   // issue early
    }

    #pragma unroll
    for (int kk = 0; kk < BK; kk += 4) {
      int kb = kk + 2 * half;
      v2f a = *(const v2f*)(&xs[cur][nlo * LSTR + kb]);
      v2f b = *(const v2f*)(&ws[cur][mat][(nsub * 16 + nlo) * LSTR + kb]);
      c = __builtin_amdgcn_wmma_f32_16x16x4_f32(false, a, false, b,
                                                (short)0, c, false, false);
    }

    if (pref)   // deferred store: x load latency hidden under WMMAs
      *(v4f*)(&xs[cur ^ 1][xr * LSTR + xc]) = xv;
  }

  // combine: h = silu(h1) * h3 (identical C layouts -> elementwise per lane)
  if (mat == 1) {
    float* dst = &hbuf[nsub * 288 + lane * 9];
    #pragma unroll
    for (int r = 0; r < 8; ++r) dst[r] = c[r];
  }
  __syncthreads();
  if (mat == 0) {
    const float* s3 = &hbuf[nsub * 288 + lane * 9];
    #pragma unroll
    for (int r = 0; r < 8; ++r) {
      float g   = c[r];
      float val = (g / (1.0f + __expf(-g))) * s3[r];
      int M = r + 8 * half;
      if (m0 + M < cnt) {
        int pid = pairs[e * MAXP + m0 + M];
        h[(size_t)pid * I + (i0 + nsub * 16 + nlo)] = val;
      }
    }
  }
}

// ---------------- GEMM2: out[pid,d] = h[pid,:] . w2[e,d,:] ----------------
// grid: (D/128, MAXP/16, E), block 256 (8 waves x one 16x16 subtile of N=128).

__global__ __launch_bounds__(256)
void moe_gemm2(const float* __restrict__ h, const float* __restrict__ w2,
               const int* __restrict__ counts, const int* __restrict__ pairs,
               float* __restrict__ out) {
  const int e   = blockIdx.z;
  const int cnt = counts[e];
  const int m0  = blockIdx.y * 16;
  if (m0 >= cnt) return;
  const int d0  = blockIdx.x * 128;

  __shared__ float hs[2][16 * LSTR];
  __shared__ float ws[2][128 * LSTR];

  const int tid  = threadIdx.x;
  const int lane = tid & 31;
  const int wave = tid >> 5;       // nsub = wave
  const int half = lane >> 4;
  const int nlo  = lane & 15;

  const int xr = tid >> 4;
  const int xc = (tid & 15) * 4;
  const float* hrow = nullptr;
  if (m0 + xr < cnt) {
    int pid = pairs[e * MAXP + m0 + xr];
    hrow = h + (size_t)pid * I;
  }

  const float* w2base = w2 + ((size_t)e * D + d0) * I;

  if (tid < 32)
    tdm_load_tile(lds_off(&ws[0][0]), w2base, 128, I);
  {
    v4f hv = {};
    if (hrow) hv = *(const v4f*)(hrow + xc);
    *(v4f*)(&hs[0][xr * LSTR + xc]) = hv;
  }

  v8f c = {};
  const int nIter = I / BK;
  for (int it = 0; it < nIter; ++it) {
    const int cur = it & 1;
    if (tid < 32) __builtin_amdgcn_s_wait_tensorcnt(0);
    __syncthreads();

    const bool pref = (it + 1 < nIter);
    v4f hv = {};
    if (pref) {
      const int k0n = (it + 1) * BK;
      if (tid < 32)
        tdm_load_tile(lds_off(&ws[cur ^ 1][0]), w2base + k0n, 128, I);
      if (hrow) hv = *(const v4f*)(hrow + k0n + xc);
    }

    #pragma unroll
    for (int kk = 0; kk < BK; kk += 4) {
      int kb = kk + 2 * half;
      v2f a = *(const v2f*)(&hs[cur][nlo * LSTR + kb]);
      v2f b = *(const v2f*)(&ws[cur][(wave * 16 + nlo) * LSTR + kb]);
      c = __builtin_amdgcn_wmma_f32_16x16x4_f32(false, a, false, b,
                                                (short)0, c, false, false);
    }

    if (pref)
      *(v4f*)(&hs[cur ^ 1][xr * LSTR + xc]) = hv;
  }

  #pragma unroll
  for (int r = 0; r < 8; ++r) {
    int M = r + 8 * half;
    if (m0 + M < cnt) {
      int pid = pairs[e * MAXP + m0 + M];
      out[(size_t)pid * D + (d0 + wave * 16 + nlo)] = c[r];
    }
  }
}

// ---------------- launch ----------------

extern "C" void kernel_launch(void* const* d_in, const int* in_sizes, int n_in,
                              void* d_out, int out_size, void* d_ws, size_t ws_size,
                              hipStream_t stream) {
  const float* x    = (const float*)d_in[0];
  const int*   eidx = (const int*)d_in[1];
  const float* w1   = (const float*)d_in[2];
  const float* w2   = (const float*)d_in[3];
  const float* w3   = (const float*)d_in[4];
  float* out = (float*)d_out;

  int*   counts = (int*)d_ws;
  int*   pairs  = counts + 16;                  // 64B-aligned
  float* h      = (float*)(pairs + E * MAXP);   // [T*A, I] fp32, 64 MB

  moe_zero<<<1, 32, 0, stream>>>(counts);
  moe_route<<<(T * A + 255) / 256, 256, 0, stream>>>(eidx, counts, pairs);
  moe_gemm1<<<dim3(I / 64, MAXP / 16, E), 256, 0, stream>>>(x, w1, w3, counts, pairs, h);
  moe_gemm2<<<dim3(D / 128, MAXP / 16, E), 256, 0, stream>>>(h, w2, counts, pairs, out);
}